// UpSample2d_23003844837862
// MI455X (gfx1250) — compile-verified
//
#include <hip/hip_runtime.h>

// UpSample2d (RATIO=2, KSIZE=12, reflect pad 5) as 4-phase 6x6 depthwise conv,
// computed with V_WMMA_F32_16X16X4_F32 via a banded-Toeplitz matmul formulation.
//
//   out[2u+ry, 2v+rx] = 4 * sum_{ty,tx<6} x[refl(u+ty+ry-3), refl(v+tx+rx-3)]
//                           * filt[11-2ty-ry, 11-2tx-rx]
//
// Per wave: one 16x16 block of (u,v) bases -> 32x32 output pixels.
//   A[m, s]  = patch[m + ty + ry, s]           (Toeplitz slab from LDS)
//   B[s, n]  = filtrow[s - n - rx] (banded)    (precomputed LDS table)
//   C(16x16) accumulates over ty (6) x K-chunks (6), K=4 per wmma.

typedef __attribute__((ext_vector_type(2))) float v2f;
typedef __attribute__((ext_vector_type(8))) float v8f;

#define HIN   256
#define WIN   256
#define HOUT  512
#define WOUT  512
#define NIMG  256            // B*C = 4*64
#define WAVES_PER_BLOCK 8
#define PATCH_ROWS 22        // rows 0..21 used
#define PATCH_COLS 24        // cols 0..23 used (last K-chunk reads s=20..23)

__device__ __forceinline__ int reflect256(int q) {
    q = (q < 0) ? -q : q;                       // single reflection suffices (q in [-3, 260])
    q = (q > HIN - 1) ? (2 * (HIN - 1) - q) : q;
    return q;
}

__global__ __launch_bounds__(256)
void upsample2d_wmma_kernel(const float* __restrict__ x,
                            const float* __restrict__ filt,
                            float* __restrict__ out) {
    __shared__ float sFilt[144];                       // 12x12
    __shared__ float sBtab[144 * 64];                  // 144 banded B mats, per-lane layout
    __shared__ float sPatch[WAVES_PER_BLOCK][PATCH_ROWS * PATCH_COLS];

    const int tid  = threadIdx.x;
    const int lane = tid & 31;
    const int wave = tid >> 5;

    // ---- stage filter into LDS ------------------------------------------------
    if (tid < 144) sFilt[tid] = filt[tid];
    __syncthreads();

    // ---- pre-bake all 144 banded B matrices (data-independent) ---------------
    // Entry e = ((ry*2+rx)*6 + ty)*6 + chunk. Stored per TARGET lane L:
    //   assumes B(4x16) layout mirrors documented A(16x4) layout with N<->M:
    //   lane L<16 holds (K=0,N=L),(K=1,N=L); lane L>=16 holds (K=2,N=L-16),(K=3,..)
    for (int idx = tid; idx < 144 * 32; idx += 256) {
        const int e  = idx >> 5;
        const int L  = idx & 31;
        const int ch = e % 6;
        const int ty = (e / 6) % 6;
        const int rx = (e / 36) % 2;
        const int ry = e / 72;
        const int n  = L & 15;
        const int k0 = (L < 16) ? 0 : 2;
        const int frow = (11 - 2 * ty - ry) * 12;
#pragma unroll
        for (int j = 0; j < 2; ++j) {
            const int s  = ch * 4 + k0 + j;
            const int tx = s - n - rx;
            const float v = (tx >= 0 && tx < 6) ? sFilt[frow + (11 - 2 * tx - rx)] : 0.0f;
            sBtab[e * 64 + L * 2 + j] = v;
        }
    }

    // ---- per-wave tile: 16x16 bases (U..U+15, X..X+15) -> 32x32 outputs -------
    const int tileId = blockIdx.x * WAVES_PER_BLOCK + wave;   // < 65536
    const int img    = tileId >> 8;                           // 256 tiles / image
    const int t      = tileId & 255;
    const int U      = (t >> 4) * 16;
    const int X      = (t & 15) * 16;

    const float* __restrict__ xin  = x   + (size_t)img * (HIN * WIN);
    float*       __restrict__ oimg = out + (size_t)img * (HOUT * WOUT);

    // stage 22x24 input patch (reflection resolved here); per-wave region, no barrier
    float* patch = &sPatch[wave][0];
    for (int i = lane; i < PATCH_ROWS * PATCH_COLS; i += 32) {
        const int r = i / PATCH_COLS;
        const int c = i - r * PATCH_COLS;
        const int q = reflect256(U + r - 3);
        const int p = reflect256(X + c - 3);
        patch[r * PATCH_COLS + c] = xin[q * WIN + p];
    }
    __syncthreads();   // sBtab visible to all waves; patch dep handled by dscnt

    // A operand addressing per documented 16x4 f32 layout:
    //   lanes 0-15:  (M=lane, K=0..1) -> cols s, s+1
    //   lanes 16-31: (M=lane-16, K=2..3) -> cols s+2, s+3
    const int  laneHi = (lane >= 16) ? 1 : 0;
    const int  mrow   = lane & 15;
    const float* aBase = patch + (laneHi ? 2 : 0);

#pragma unroll
    for (int ry = 0; ry < 2; ++ry) {
        v8f c0 = {};   // phase rx=0 accumulator
        v8f c1 = {};   // phase rx=1 accumulator
#pragma unroll
        for (int ty = 0; ty < 6; ++ty) {
            const float* ap  = aBase + (mrow + ty + ry) * PATCH_COLS;
            const float* b0p = sBtab + (size_t)(((ry * 2 + 0) * 6 + ty) * 6) * 64 + lane * 2;
            const float* b1p = sBtab + (size_t)(((ry * 2 + 1) * 6 + ty) * 6) * 64 + lane * 2;
#pragma unroll
            for (int ch = 0; ch < 6; ++ch) {
                v2f a  = *(const v2f*)(ap  + ch * 4);
                v2f b0 = *(const v2f*)(b0p + ch * 64);
                v2f b1 = *(const v2f*)(b1p + ch * 64);
                c0 = __builtin_amdgcn_wmma_f32_16x16x4_f32(
                         false, a, false, b0, (short)0, c0, false, false);
                c1 = __builtin_amdgcn_wmma_f32_16x16x4_f32(
                         false, a, false, b1, (short)0, c1, false, false);
            }
        }
        // Interleave rx=0/rx=1 into float2 -> contiguous 8B/lane, coalesced 128B.
        // C layout: VGPR v -> row (v + 8*laneHi), col (lane&15).
        const int n = lane & 15;
#pragma unroll
        for (int v = 0; v < 8; ++v) {
            const int m  = v + (laneHi ? 8 : 0);
            const int oy = 2 * (U + m) + ry;
            const int ox = 2 * (X + n);
            float2 val = make_float2(4.0f * c0[v], 4.0f * c1[v]);
            *(float2*)(oimg + (size_t)oy * WOUT + ox) = val;
        }
    }
}

extern "C" void kernel_launch(void* const* d_in, const int* in_sizes, int n_in,
                              void* d_out, int out_size, void* d_ws, size_t ws_size,
                              hipStream_t stream) {
    (void)in_sizes; (void)n_in; (void)out_size; (void)d_ws; (void)ws_size;
    const float* x    = (const float*)d_in[0];   // (4,64,256,256) fp32
    const float* filt = (const float*)d_in[1];   // (12,12) fp32
    float* out        = (float*)d_out;           // (4,64,512,512) fp32

    // 256 images * 256 tiles / 8 waves-per-block = 8192 blocks of 256 threads
    const int totalTiles = NIMG * 256;
    dim3 grid(totalTiles / WAVES_PER_BLOCK);
    upsample2d_wmma_kernel<<<grid, 256, 0, stream>>>(x, filt, out);
}